// FeedForwardNet_65867618451897
// MI455X (gfx1250) — compile-verified
//
#include <hip/hip_runtime.h>

// Problem constants (from the reference)
#define B_     4096
#define NIN_   512
#define L_     16
#define N_     512
#define K_     32
#define NOUT_  256
#define TOTAL_ (NIN_ + L_ * N_)   // 8704 columns in buf

typedef float v2f __attribute__((ext_vector_type(2)));
typedef float v8f __attribute__((ext_vector_type(8)));

// ---------------------------------------------------------------------------
// buf[:, :NIN] = x   (row-major buf: B x TOTAL)
// ---------------------------------------------------------------------------
__global__ __launch_bounds__(256)
void ffn_init_buf(const float* __restrict__ x, float* __restrict__ buf) {
    int i = blockIdx.x * blockDim.x + threadIdx.x;   // over B_*NIN_
    int b = i / NIN_;
    int f = i - b * NIN_;
    buf[(size_t)b * TOTAL_ + f] = x[i];
}

// ---------------------------------------------------------------------------
// One sparse layer: out[b, n] = sigmoid( sum_k buf[b, idx[n,k]] * W[n,k] + bias[n] )
// One wave (32 lanes) computes a 16-batch x 16-neuron tile with 128 chained
// V_WMMA_F32_16X16X4_F32 ops (exact f32 accumulation; block-diagonal B).
//
// Layouts (CDNA5 ISA):
//   A (16x4 f32, 2 VGPRs): lane<16 -> M=lane, K=VGPR;  lane>=16 -> M=lane-16, K=VGPR+2
//   B (4x16 f32, 2 VGPRs): col = lane&15, K = 2*(lane>=16) + VGPR
//   C/D (16x16 f32, 8 VGPRs): VGPR r -> M = r + 8*(lane>=16), N = lane&15
// ---------------------------------------------------------------------------
__device__ __forceinline__ v8f ffn_tile_accum(const float* __restrict__ buf,
                                              const float* __restrict__ W,
                                              const int*   __restrict__ idx,
                                              int n0, size_t rowA, int hi, int lo) {
    v8f acc = {};
    #pragma unroll 1
    for (int j = 0; j < 16; ++j) {
        const int nj = n0 + j;
        const int*   idr = idx + nj * K_;
        const float* wr  = W   + nj * K_;
        const bool mine = (lo == j);
        #pragma unroll
        for (int kk = 0; kk < 8; ++kk) {
            const int kb = kk * 4 + 2 * hi;          // even -> 8B aligned pairs
            const int2   fp = *(const int2*)  (idr + kb);
            const float2 wp = *(const float2*)(wr  + kb);
            v2f a, bm;
            a.x = buf[rowA + (size_t)fp.x];
            a.y = buf[rowA + (size_t)fp.y];
            bm.x = mine ? wp.x : 0.0f;               // only column j nonzero
            bm.y = mine ? wp.y : 0.0f;
            acc = __builtin_amdgcn_wmma_f32_16x16x4_f32(
                      false, a, false, bm, (short)0, acc, false, false);
        }
    }
    return acc;
}

__global__ __launch_bounds__(32)
void ffn_layer_kernel(const float* __restrict__ buf_in,
                      float*       __restrict__ buf,
                      const float* __restrict__ W,     // N x K (this layer)
                      const float* __restrict__ bias,  // N
                      const int*   __restrict__ idx,   // N x K
                      int out_off) {                   // NIN + l*N
    const int lane = threadIdx.x;
    const int hi = lane >> 4;
    const int lo = lane & 15;
    const int bt = blockIdx.x;                 // batch tile (16 rows)
    const int n0 = blockIdx.y * 16;            // neuron tile (16 cols)
    const size_t rowA = (size_t)(bt * 16 + lo) * TOTAL_;

    v8f acc = ffn_tile_accum(buf_in, W, idx, n0, rowA, hi, lo);

    const float bv = bias[n0 + lo];
    #pragma unroll
    for (int r = 0; r < 8; ++r) {
        float v = acc[r] + bv;
        float o = 1.0f / (1.0f + __expf(-v));
        buf[(size_t)(bt * 16 + r + 8 * hi) * TOTAL_ + out_off + n0 + lo] = o;
    }
}

__global__ __launch_bounds__(32)
void ffn_out_kernel(const float* __restrict__ buf,
                    const float* __restrict__ W,     // NOUT x K
                    const float* __restrict__ bias,  // NOUT
                    const int*   __restrict__ idx,   // NOUT x K
                    float*       __restrict__ out) { // B x NOUT
    const int lane = threadIdx.x;
    const int hi = lane >> 4;
    const int lo = lane & 15;
    const int bt = blockIdx.x;
    const int n0 = blockIdx.y * 16;
    const size_t rowA = (size_t)(bt * 16 + lo) * TOTAL_;

    v8f acc = ffn_tile_accum(buf, W, idx, n0, rowA, hi, lo);

    const float bv = bias[n0 + lo];
    #pragma unroll
    for (int r = 0; r < 8; ++r) {
        float v = acc[r] + bv;
        float o = 1.0f / (1.0f + __expf(-v));
        out[(size_t)(bt * 16 + r + 8 * hi) * NOUT_ + n0 + lo] = o;
    }
}

// ---------------------------------------------------------------------------
// Inputs (setup_inputs order): x, W, b, W_out, b_out, idx, idx_out
// Workspace: buf (B x TOTAL f32 = 142.6 MB) lives in d_ws -> stays L2-resident.
// ---------------------------------------------------------------------------
extern "C" void kernel_launch(void* const* d_in, const int* in_sizes, int n_in,
                              void* d_out, int out_size, void* d_ws, size_t ws_size,
                              hipStream_t stream) {
    const float* x     = (const float*)d_in[0];
    const float* W     = (const float*)d_in[1];   // L*N*K
    const float* b     = (const float*)d_in[2];   // L*N
    const float* W_out = (const float*)d_in[3];   // NOUT*K
    const float* b_out = (const float*)d_in[4];   // NOUT
    const int*   idx     = (const int*)d_in[5];   // L*N*K
    const int*   idx_out = (const int*)d_in[6];   // NOUT*K

    float* buf = (float*)d_ws;                    // B x TOTAL
    float* out = (float*)d_out;                   // B x NOUT

    ffn_init_buf<<<(B_ * NIN_) / 256, 256, 0, stream>>>(x, buf);

    dim3 grid(B_ / 16, N_ / 16);
    for (int l = 0; l < L_; ++l) {
        ffn_layer_kernel<<<grid, 32, 0, stream>>>(
            buf, buf,
            W   + (size_t)l * N_ * K_,
            b   + (size_t)l * N_,
            idx + (size_t)l * N_ * K_,
            NIN_ + l * N_);
    }

    dim3 grid_out(B_ / 16, NOUT_ / 16);
    ffn_out_kernel<<<grid_out, 32, 0, stream>>>(buf, W_out, b_out, idx_out, out);
}